// CapsuleNetwork_26637387170318
// MI455X (gfx1250) — compile-verified
//
#include <hip/hip_runtime.h>

// CDNA5 / gfx1250 — wave32, WMMA 16x16x4 fp32.
typedef float v2f __attribute__((ext_vector_type(2)));
typedef float v8f __attribute__((ext_vector_type(8)));

static __device__ __forceinline__ v8f wmma_f32(v2f a, v2f b, v8f c) {
    // D = A(16x4) * B(4x16) + C(16x16), all fp32.
    return __builtin_amdgcn_wmma_f32_16x16x4_f32(false, a, false, b, (short)0, c, false, false);
}

// ------------- conv1 weight transpose: w[oc][j] -> w1T[j][oc] --------------
__global__ __launch_bounds__(256) void w1trans_kernel(const float* __restrict__ w,
                                                      float* __restrict__ w1T) {
    int gid = blockIdx.x * 256 + threadIdx.x;   // 81*256 = 20736 exact
    int oc = gid % 256, j = gid / 256;
    w1T[gid] = w[oc * 81 + j];
}

// ---------------- conv1: x[256,1,28,28] -> hT[b][y][x][oc] (NHWC), ReLU ----
__global__ __launch_bounds__(256) void conv1_kernel(const float* __restrict__ x,
                                                    const float* __restrict__ w1T,
                                                    const float* __restrict__ bias,
                                                    float* __restrict__ hT) {
    __shared__ float patch[81];
    int bid = blockIdx.x;                 // b*400 + y*20 + xx
    int b = bid / 400;
    int r = bid % 400;
    int y = r / 20, xx = r % 20;
    int t = threadIdx.x;                  // t = output channel
    if (t < 81) {
        int i = t / 9, j = t % 9;
        patch[t] = x[b * 784 + (y + i) * 28 + (xx + j)];
    }
    __syncthreads();
    float acc = bias[t];
#pragma unroll 9
    for (int j = 0; j < 81; ++j) acc = fmaf(patch[j], w1T[j * 256 + t], acc);
    hT[bid * 256 + t] = fmaxf(acc, 0.0f);
}

// --- caps weight transpose+interleave: caps_w[n][ic][s81] ->
//     Wt[s81][ic/4][n][ic%4]  (so a lane's B-fragment pair is one b64) ------
__global__ __launch_bounds__(256) void wtrans_kernel(const float* __restrict__ cw,
                                                     float* __restrict__ Wt) {
    int gid = blockIdx.x * 256 + threadIdx.x;   // 81*256*256 = 5,308,416 exact
    int ic4 = gid & 3;
    int n   = (gid >> 2) & 255;
    int icq = (gid >> 10) & 63;
    int s81 = gid >> 16;
    int ic  = icq * 4 + ic4;
    Wt[gid] = cw[(n * 256 + ic) * 81 + s81];
}

// --- caps conv implicit GEMM: S[b][sp][n] = sum_{ky,kx,ic} hT * Wt (+bias) --
// M = b*36+sp (9216), N = 256, K = 81*256. Each wave computes a 32x32 output
// region (2x2 WMMA tiles, 4 accumulators): 4 b64 loads feed 4 WMMAs.
// Block = 8 waves arranged 4(M) x 2(N) -> 128x64 block tile. grid = (72, 4).
__global__ __launch_bounds__(256) void caps_gemm(const float* __restrict__ hT,
                                                 const float* __restrict__ Wt,
                                                 const float* __restrict__ cb,
                                                 float* __restrict__ S) {
    int tid  = threadIdx.x;
    int w    = tid >> 5;
    int lane = tid & 31;
    int hi   = lane >> 4;        // K offset +2 for upper lane half
    int lm   = lane & 15;
    int mbase = blockIdx.x * 128 + (w >> 1) * 32;
    int nbase = blockIdx.y * 64 + (w & 1) * 32;

    // Two A rows per lane (m-tiles 0 and 1)
    int m0 = mbase + lm, m1 = mbase + 16 + lm;
    int b0 = m0 / 36, sp0 = m0 % 36;
    int b1 = m1 / 36, sp1 = m1 % 36;
    const float* aptr0 = hT + ((b0 * 20 + 2 * (sp0 / 6)) * 20 + 2 * (sp0 % 6)) * 256 + 2 * hi;
    const float* aptr1 = hT + ((b1 * 20 + 2 * (sp1 / 6)) * 20 + 2 * (sp1 % 6)) * 256 + 2 * hi;

    v8f acc00 = {}, acc01 = {}, acc10 = {}, acc11 = {};
#pragma unroll 1
    for (int ky = 0; ky < 9; ++ky) {
#pragma unroll 1
        for (int kx = 0; kx < 9; ++kx) {
            int aoff = (ky * 20 + kx) * 256;
            const float* arow0 = aptr0 + aoff;
            const float* arow1 = aptr1 + aoff;
            // Wt element (s81, ic, n) lives at s81*65536 + (ic>>2)*1024 + n*4 + (ic&3)
            const float* bcol0 = Wt + (ky * 9 + kx) * 65536 + (nbase + lm) * 4 + 2 * hi;
            const float* bcol1 = bcol0 + 64;   // +16 columns
#pragma unroll 4
            for (int ic = 0; ic < 256; ic += 4) {
                v2f a0 = *(const v2f*)(arow0 + ic);
                v2f a1 = *(const v2f*)(arow1 + ic);
                v2f bq0 = *(const v2f*)(bcol0 + ic * 256);  // {K=ic+2hi, ic+2hi+1}
                v2f bq1 = *(const v2f*)(bcol1 + ic * 256);
                acc00 = wmma_f32(a0, bq0, acc00);
                acc01 = wmma_f32(a0, bq1, acc01);
                acc10 = wmma_f32(a1, bq0, acc10);
                acc11 = wmma_f32(a1, bq1, acc11);
            }
        }
    }
#pragma unroll
    for (int r = 0; r < 8; ++r) {
        int mo0 = mbase + r + 8 * hi;
        int mo1 = mbase + 16 + r + 8 * hi;
        int no0 = nbase + lm;
        int no1 = nbase + 16 + lm;
        S[mo0 * 256 + no0] = acc00[r] + cb[no0];
        S[mo0 * 256 + no1] = acc01[r] + cb[no1];
        S[mo1 * 256 + no0] = acc10[r] + cb[no0];
        S[mo1 * 256 + no1] = acc11[r] + cb[no1];
    }
}

// ------------- n2[prim] = sqrt(sum over b,sp,d of S^2), two-stage ----------
__global__ __launch_bounds__(256) void n2_stage1(const float* __restrict__ S,
                                                 float* __restrict__ part) {
    __shared__ float red[256];
    int b = blockIdx.x, n = threadIdx.x;
    const float* p = S + b * 36 * 256 + n;
    float s = 0.0f;
    for (int sp = 0; sp < 36; ++sp) { float v = p[sp * 256]; s = fmaf(v, v, s); }
    red[n] = s;
    __syncthreads();
    if (n < 32) {
        float t = 0.0f;
        for (int d = 0; d < 8; ++d) t += red[n * 8 + d];
        part[b * 32 + n] = t;
    }
}

__global__ void n2_stage2(const float* __restrict__ part, float* __restrict__ n2) {
    int p = threadIdx.x;
    if (p < 32) {
        float s = 0.0f;
        for (int b = 0; b < 256; ++b) s += part[b * 32 + p];
        n2[p] = sqrtf(s);
    }
}

// --- squash: vs[b][i=sp*32+prim][g=d] = (n2/(1+n2)) * S / L1_d(|S|) --------
__global__ __launch_bounds__(256) void vs_kernel(const float* __restrict__ S,
                                                 const float* __restrict__ n2,
                                                 float* __restrict__ vs) {
    int gid  = blockIdx.x * 256 + threadIdx.x;   // 294912 exact
    int b    = gid / 1152;
    int rem  = gid % 1152;
    int sp   = rem >> 5, prim = rem & 31;
    const float* sp8 = S + (b * 36 + sp) * 256 + prim * 8;
    float v[8], n1 = 0.0f;
#pragma unroll
    for (int d = 0; d < 8; ++d) { v[d] = sp8[d]; n1 += fabsf(v[d]); }
    float nn = n2[prim];
    float sc = nn / (1.0f + nn) / n1;
    float* out = vs + gid * 8;
#pragma unroll
    for (int d = 0; d < 8; ++d) out[d] = v[d] * sc;
}

// ------------- zero routing logits -----------------------------------------
__global__ __launch_bounds__(256) void zero_bl(float* __restrict__ Bl) {
    Bl[blockIdx.x * 256 + threadIdx.x] = 0.0f;   // 45*256 = 11520 exact
}

// ------------- softmax over in_caps (1152) per out_cap ---------------------
__global__ __launch_bounds__(256) void softmax_kernel(const float* __restrict__ Bl,
                                                      float* __restrict__ c) {
    __shared__ float red[256];
    int o = blockIdx.x, t = threadIdx.x;
    const float* row = Bl + o * 1152;
    float m = -1e30f;
    for (int j = t; j < 1152; j += 256) m = fmaxf(m, row[j]);
    red[t] = m; __syncthreads();
    for (int off = 128; off > 0; off >>= 1) { if (t < off) red[t] = fmaxf(red[t], red[t + off]); __syncthreads(); }
    float mx = red[0]; __syncthreads();
    float s = 0.0f;
    for (int j = t; j < 1152; j += 256) s += expf(row[j] - mx);
    red[t] = s; __syncthreads();
    for (int off = 128; off > 0; off >>= 1) { if (t < off) red[t] += red[t + off]; __syncthreads(); }
    float inv = 1.0f / red[0];
    for (int j = t; j < 1152; j += 256) c[o * 1152 + j] = expf(row[j] - mx) * inv;
}

// --- sdig[b][o][f] = sum_{i,g} vs[b,i,g] * (c[o,i]*W[o,i,g,f]), WMMA -------
// Per o: GEMM M=256(b) x N=16(f) x K=9216(i*8+g). 160 wave-tiles, 20 blocks.
// All 8 waves of a block share one o, so c[o][*] is cached in LDS.
__global__ __launch_bounds__(256) void sdig_gemm(const float* __restrict__ vs,
                                                 const float* __restrict__ dw,
                                                 const float* __restrict__ c,
                                                 float* __restrict__ sdig) {
    __shared__ float cs[1152];
    int tid = threadIdx.x;
    int o = blockIdx.x >> 1;
    for (int j = tid; j < 1152; j += 256) cs[j] = c[o * 1152 + j];
    __syncthreads();
    int w = tid >> 5, lane = tid & 31;
    int hi = lane >> 4, lm = lane & 15;
    int m_w = (((blockIdx.x & 1) << 3) + w) * 16;     // b-tile
    const float* arow = vs + (m_w + lm) * 9216 + 2 * hi;
    const float* brow = dw + o * (9216 * 16) + 2 * hi * 16 + lm;
    v8f acc = {};
#pragma unroll 2
    for (int k = 0; k < 9216; k += 4) {
        int k0 = k + 2 * hi;
        v2f a = *(const v2f*)(arow + k);
        v2f bf;
        bf.x = brow[k * 16]      * cs[k0 >> 3];
        bf.y = brow[k * 16 + 16] * cs[(k0 + 1) >> 3];
        acc = wmma_f32(a, bf, acc);
    }
#pragma unroll
    for (int r = 0; r < 8; ++r)
        sdig[(m_w + r + 8 * hi) * 160 + o * 16 + lm] = acc[r];
}

// ------------- sn2 = sqrt(sum of all sdig^2), single block, deterministic ---
__global__ __launch_bounds__(256) void sn2_kernel(const float* __restrict__ sdig,
                                                  float* __restrict__ sn2) {
    __shared__ float red[256];
    int t = threadIdx.x;
    float s = 0.0f;
    for (int j = t; j < 40960; j += 256) { float v = sdig[j]; s = fmaf(v, v, s); }
    red[t] = s; __syncthreads();
    for (int off = 128; off > 0; off >>= 1) { if (t < off) red[t] += red[t + off]; __syncthreads(); }
    if (t == 0) *sn2 = sqrtf(red[0]);
}

// ------------- vout = (sn2/(1+sn2)) * sdig / sum_o |sdig| ------------------
__global__ __launch_bounds__(256) void vout_kernel(const float* __restrict__ sdig,
                                                   const float* __restrict__ sn2p,
                                                   float* __restrict__ vout) {
    int gid = blockIdx.x * 256 + threadIdx.x;   // 4096 exact
    int b = gid >> 4, f = gid & 15;
    float sn1 = 0.0f;
    for (int o = 0; o < 10; ++o) sn1 += fabsf(sdig[b * 160 + o * 16 + f]);
    float sn2 = *sn2p;
    float sc = sn2 / (1.0f + sn2) / sn1;
    for (int o = 0; o < 10; ++o) vout[b * 160 + o * 16 + f] = sdig[b * 160 + o * 16 + f] * sc;
}

// --- Q[(o*16+f)][(i*8+g)] = sum_b vout[b,o,f]*vs[b,i,g], WMMA --------------
// GEMM M=160 x N=9216 x K=256; 5760 wave-tiles -> 720 blocks.
__global__ __launch_bounds__(256) void q_gemm(const float* __restrict__ vout,
                                              const float* __restrict__ vs,
                                              float* __restrict__ Q) {
    int tid = threadIdx.x;
    int w = tid >> 5, lane = tid & 31;
    int hi = lane >> 4, lm = lane & 15;
    int gw = blockIdx.x * 8 + w;
    int m_w = (gw % 10) * 16;
    int n_w = (gw / 10) * 16;
    v8f acc = {};
#pragma unroll 4
    for (int k = 0; k < 256; k += 4) {
        int k0 = k + 2 * hi;
        v2f a, bf;
        a.x  = vout[k0 * 160 + m_w + lm];
        a.y  = vout[(k0 + 1) * 160 + m_w + lm];
        bf.x = vs[k0 * 9216 + n_w + lm];
        bf.y = vs[(k0 + 1) * 9216 + n_w + lm];
        acc = wmma_f32(a, bf, acc);
    }
#pragma unroll
    for (int r = 0; r < 8; ++r)
        Q[(m_w + r + 8 * hi) * 9216 + n_w + lm] = acc[r];
}

// ------------- Bl[o,i] += (1/B) * sum_{g,f} W[o,i,g,f] * Q[o*16+f][i*8+g] --
__global__ __launch_bounds__(256) void bl_update(const float* __restrict__ dw,
                                                 const float* __restrict__ Q,
                                                 float* __restrict__ Bl) {
    int gid = blockIdx.x * 256 + threadIdx.x;   // 11520 exact
    int o = gid / 1152, i = gid % 1152;
    const float* wp = dw + (o * 1152 + i) * 128;
    float s = 0.0f;
#pragma unroll
    for (int g = 0; g < 8; ++g)
        for (int f = 0; f < 16; ++f)
            s = fmaf(wp[g * 16 + f], Q[(o * 16 + f) * 9216 + i * 8 + g], s);
    Bl[gid] += s * (1.0f / 256.0f);
}

// ------------- classes[b][o] = ||vout[b,o,:]|| -----------------------------
__global__ __launch_bounds__(256) void classes_kernel(const float* __restrict__ vout,
                                                      float* __restrict__ cls) {
    int gid = blockIdx.x * 256 + threadIdx.x;   // 2560 exact
    int b = gid / 10, o = gid % 10;
    float s = 0.0f;
    for (int f = 0; f < 16; ++f) { float v = vout[b * 160 + o * 16 + f]; s = fmaf(v, v, s); }
    cls[gid] = sqrtf(s);
}

extern "C" void kernel_launch(void* const* d_in, const int* in_sizes, int n_in,
                              void* d_out, int out_size, void* d_ws, size_t ws_size,
                              hipStream_t stream) {
    (void)in_sizes; (void)n_in; (void)out_size; (void)ws_size;
    const float* x   = (const float*)d_in[0];
    const float* c1w = (const float*)d_in[1];
    const float* c1b = (const float*)d_in[2];
    const float* cw  = (const float*)d_in[3];
    const float* cb  = (const float*)d_in[4];
    const float* dw  = (const float*)d_in[5];

    float* out  = (float*)d_out;
    float* cls  = out;            // [256,10]
    float* vout = out + 2560;     // [256,10,16] — final routing iter leaves it correct

    float* ws     = (float*)d_ws;
    float* hT     = ws;                    // 26,214,400  (NHWC conv1 out)
    float* Wt     = hT + 26214400;         //  5,308,416  (interleaved caps weights)
    float* S      = Wt + 5308416;          //  2,359,296  (caps conv out, [b][sp][n])
    float* vs     = S  + 2359296;          //  2,359,296  ([b][1152][8])
    float* Q      = vs + 2359296;          //  1,474,560  ([160][9216])
    float* Bl     = Q  + 1474560;          //     11,520
    float* cc     = Bl + 11520;            //     11,520
    float* sdig   = cc + 11520;            //     40,960
    float* n2part = sdig + 40960;          //      8,192
    float* n2     = n2part + 8192;         //         32
    float* sn2    = n2 + 32;               //          1
    float* w1T    = sn2 + 1;               //     20,736
    // total ~37.8M floats (~151 MB)

    zero_bl<<<45, 256, 0, stream>>>(Bl);
    w1trans_kernel<<<81, 256, 0, stream>>>(c1w, w1T);
    conv1_kernel<<<102400, 256, 0, stream>>>(x, w1T, c1b, hT);
    wtrans_kernel<<<20736, 256, 0, stream>>>(cw, Wt);
    caps_gemm<<<dim3(72, 4), 256, 0, stream>>>(hT, Wt, cb, S);
    n2_stage1<<<256, 256, 0, stream>>>(S, n2part);
    n2_stage2<<<1, 32, 0, stream>>>(n2part, n2);
    vs_kernel<<<1152, 256, 0, stream>>>(S, n2, vs);

    for (int it = 0; it < 3; ++it) {
        softmax_kernel<<<10, 256, 0, stream>>>(Bl, cc);
        sdig_gemm<<<20, 256, 0, stream>>>(vs, dw, cc, sdig);
        sn2_kernel<<<1, 256, 0, stream>>>(sdig, sn2);
        vout_kernel<<<16, 256, 0, stream>>>(sdig, sn2, vout);
        if (it < 2) {   // Bl is unused after the last iteration
            q_gemm<<<720, 256, 0, stream>>>(vout, vs, Q);
            bl_update<<<45, 256, 0, stream>>>(dw, Q, Bl);
        }
    }
    classes_kernel<<<10, 256, 0, stream>>>(vout, cls);
}